// Masking_89326729822839
// MI455X (gfx1250) — compile-verified
//
#include <hip/hip_runtime.h>
#include <math.h>

// ---------------- problem constants (from reference: B=16, T=4096, F=512) ---
#define BROWS       16
#define NPERROW     (4096 * 512)            // 2,097,152 elements per row (2^21)
#define N4PERROW    (NPERROW / 4)           // 524,288 float4 per row (2^19)
#define HBINS       4096                    // per-row histogram stride (max bins)

#define HIST_BLK    256
#define HIST_WAVES  (HIST_BLK / 32)
#define HIST_SLICES 64
#define HIST_ITERS  (N4PERROW / (HIST_SLICES * HIST_BLK))   // 32 chunks/wave

#define AP_BLK      256

typedef float v4f __attribute__((ext_vector_type(4)));

// Monotone map: float total order -> unsigned order (and back)
__device__ __forceinline__ unsigned keyOf(float f) {
  unsigned u = __float_as_uint(f);
  return (u & 0x80000000u) ? ~u : (u | 0x80000000u);
}
__device__ __forceinline__ float keyInv(unsigned k) {
  unsigned u = (k & 0x80000000u) ? (k & 0x7fffffffu) : ~k;
  return __uint_as_float(u);
}

// ---- CDNA5 async path: global -> LDS bulk copy tracked by ASYNCcnt ----
__device__ __forceinline__ void async_load_v4(unsigned lds_addr, const v4f* gp) {
  asm volatile("global_load_async_to_lds_b128 %0, %1, off"
               :: "v"(lds_addr), "v"(gp) : "memory");
}
__device__ __forceinline__ void wait_async0() {
  asm volatile("s_wait_asynccnt 0" ::: "memory");
}
__device__ __forceinline__ void wait_async1() {
  asm volatile("s_wait_asynccnt 1" ::: "memory");
}

// ---------------- workspace layout ----------------
// [0,128)    double rowSum[16]
// [128,256)  double maskedSum[16]
// [256,320)  unsigned sel[16]
// [320,384)  unsigned kRem[16]
// [384,448)  float thr[16]
// [512, +256K)        unsigned gcnt[16][4096]
// [512+256K, +256K)   float    gsum[16][4096]
#define WS_GCNT_OFF 512
#define WS_GSUM_OFF (WS_GCNT_OFF + BROWS * HBINS * 4)
#define WS_WORDS    ((WS_GSUM_OFF + BROWS * HBINS * 4) / 4)

__global__ void k_zero(unsigned* __restrict__ w, int n) {
  int i = blockIdx.x * blockDim.x + threadIdx.x;
  if (i < n) w[i] = 0u;
}

// One radix pass: per-bin counts (ds_add_u32) AND per-bin value sums
// (ds_add_f32). Data is streamed global->LDS with double-buffered async
// copies (ASYNCcnt), then consumed from LDS.
__global__ __launch_bounds__(HIST_BLK)
void k_hist(const float* __restrict__ x, unsigned* __restrict__ gcnt,
            float* __restrict__ gsum, const unsigned* __restrict__ sel,
            unsigned cmpMask, int binShift, unsigned binMask)
{
  __shared__ unsigned shc[HBINS];
  __shared__ float    shs[HBINS];
  __shared__ v4f      stage[HIST_WAVES][2][32];
  for (int j = threadIdx.x; j < HBINS; j += HIST_BLK) { shc[j] = 0u; shs[j] = 0.0f; }
  __syncthreads();

  const int row  = blockIdx.y;
  const int w    = threadIdx.x >> 5;
  const int lane = threadIdx.x & 31;
  const unsigned selv = cmpMask ? sel[row] : 0u;

  const v4f* xr = (const v4f*)x + (size_t)row * N4PERROW
                + (size_t)blockIdx.x * (HIST_BLK * HIST_ITERS)
                + (size_t)w * (32 * HIST_ITERS);        // per-wave contiguous

  unsigned lds0 = (unsigned)(size_t)&stage[w][0][lane];
  unsigned lds1 = (unsigned)(size_t)&stage[w][1][lane];

  async_load_v4(lds0, xr + lane);                       // prologue
  for (int c = 0; c < HIST_ITERS; ++c) {
    if (c + 1 < HIST_ITERS) {
      async_load_v4((c & 1) ? lds0 : lds1, xr + (size_t)(c + 1) * 32 + lane);
      wait_async1();                                    // chunk c has landed
    } else {
      wait_async0();
    }
    v4f v = stage[w][c & 1][lane];
    float f[4] = {v.x, v.y, v.z, v.w};
#pragma unroll
    for (int q = 0; q < 4; ++q) {
      unsigned key = keyOf(f[q]);
      if ((key & cmpMask) == selv) {
        unsigned b = (key >> binShift) & binMask;
        atomicAdd(&shc[b], 1u);       // ds_add_u32
        atomicAdd(&shs[b], f[q]);     // ds_add_f32
      }
    }
  }
  __syncthreads();
  unsigned* gc = gcnt + (size_t)row * HBINS;
  float*    gs = gsum + (size_t)row * HBINS;
  for (int j = threadIdx.x; j < HBINS; j += HIST_BLK) {
    unsigned cnt = shc[j];
    if (cnt) { atomicAdd(&gc[j], cnt); atomicAdd(&gs[j], shs[j]); }
  }
}

// Per-row scan: locate rank-k bin, accumulate selected key bits, and fold the
// suffix of per-bin sums into maskedSum (level<2: bins>b; level2: bins>=b).
// Level 0 also records the full row sum. Zeroes the histograms for reuse.
__global__ __launch_bounds__(256)
void k_scan(unsigned* __restrict__ gcnt, float* __restrict__ gsum,
            const float* __restrict__ probs, unsigned* __restrict__ sel,
            unsigned* __restrict__ kRem, float* __restrict__ thr,
            double* __restrict__ rowSum, double* __restrict__ maskedSum,
            int bins, int binShift, int level)
{
  const int row = blockIdx.x;
  unsigned* hc = gcnt + (size_t)row * HBINS;
  float*    hs = gsum + (size_t)row * HBINS;
  const int t = threadIdx.x;
  const int chunk = bins >> 8;     // 16 (4096 bins) or 1 (256 bins)

  unsigned k;
  if (level == 0) {
    float p = probs[row];
    int ki = (int)ceilf((float)NPERROW * p) - 1;     // matches jnp.ceil(f32)
    k = (unsigned)(ki < 0 ? 0 : ki);
  } else {
    k = kRem[row];
  }

  unsigned pcnt = 0; double psum = 0.0;
  for (int j = 0; j < chunk; ++j) {
    pcnt += hc[t * chunk + j];
    psum += (double)hs[t * chunk + j];
  }
  __shared__ unsigned pc[256];
  __shared__ double   pd[256];
  pc[t] = pcnt; pd[t] = psum; __syncthreads();
  for (int off = 1; off < 256; off <<= 1) {
    unsigned vc = (t >= off) ? pc[t - off] : 0u;
    double   vd = (t >= off) ? pd[t - off] : 0.0;
    __syncthreads();
    pc[t] += vc; pd[t] += vd;
    __syncthreads();
  }
  const unsigned incl = pc[t], excl = incl - pcnt;
  const double exclSum  = pd[t] - psum;
  const double totalSum = pd[255];                   // valid after final barrier

  if (excl <= k && k < incl) {                       // exactly one thread
    unsigned cum = excl; double csum = exclSum;
    for (int j = 0; j < chunk; ++j) {
      unsigned c  = hc[t * chunk + j];
      double   bs = (double)hs[t * chunk + j];
      if (k < cum + c) {
        unsigned bin = (unsigned)(t * chunk + j);
        unsigned s = (level == 0) ? 0u : sel[row];
        s |= bin << binShift;
        sel[row]  = s;
        kRem[row] = k - cum;
        double kept = (level < 2) ? (totalSum - (csum + bs))   // bins > bin
                                  : (totalSum - csum);         // bins >= bin
        maskedSum[row] += kept;        // single writer, sequential launches
        if (level == 0) rowSum[row] = totalSum;
        if (level == 2) thr[row] = keyInv(s);        // exact k-th smallest
        break;
      }
      cum += c; csum += bs;
    }
  }
  __syncthreads();
  for (int j = t; j < HBINS; j += 256) { hc[j] = 0u; hs[j] = 0.0f; }
}

// out = (x >= thr) ? x * |num/den| : 0  (divide_no_nan); NT stores keep x in L2.
__global__ __launch_bounds__(AP_BLK)
void k_apply(const float* __restrict__ x, const float* __restrict__ thr,
             const double* __restrict__ rowSum, const double* __restrict__ maskedSum,
             float* __restrict__ out)
{
  size_t i4 = (size_t)blockIdx.x * AP_BLK + threadIdx.x;
  int row = (int)(i4 >> 19);                         // N4PERROW == 2^19
  float t = thr[row];
  double den = maskedSum[row];
  float scale = (den == 0.0) ? 0.0f : (float)fabs(rowSum[row] / den);
  v4f v = ((const v4f*)x)[i4];
  v4f o;
  o.x = (v.x >= t) ? v.x * scale : 0.0f;
  o.y = (v.y >= t) ? v.y * scale : 0.0f;
  o.z = (v.z >= t) ? v.z * scale : 0.0f;
  o.w = (v.w >= t) ? v.w * scale : 0.0f;
  __builtin_nontemporal_store(o, (v4f*)out + i4);
}

extern "C" void kernel_launch(void* const* d_in, const int* in_sizes, int n_in,
                              void* d_out, int out_size, void* d_ws, size_t ws_size,
                              hipStream_t stream)
{
  (void)in_sizes; (void)n_in; (void)out_size; (void)ws_size;
  const float* x     = (const float*)d_in[0];
  const float* probs = (const float*)d_in[1];
  float*       out   = (float*)d_out;

  double*   rowSum    = (double*)d_ws;
  double*   maskedSum = rowSum + BROWS;
  unsigned* sel       = (unsigned*)(maskedSum + BROWS);
  unsigned* kRem      = sel + BROWS;
  float*    thr       = (float*)(kRem + BROWS);
  unsigned* gcnt      = (unsigned*)((char*)d_ws + WS_GCNT_OFF);
  float*    gsum      = (float*)((char*)d_ws + WS_GSUM_OFF);

  // 0) zero scratch (hists + accumulators) — fresh every call, replay-safe
  k_zero<<<(WS_WORDS + 255) / 256, 256, 0, stream>>>((unsigned*)d_ws, WS_WORDS);

  dim3 hg(HIST_SLICES, BROWS);
  // level 0: key bits [31:20]  (the one true HBM pass; later passes hit L2)
  k_hist<<<hg, HIST_BLK, 0, stream>>>(x, gcnt, gsum, sel, 0u, 20, 0xFFFu);
  k_scan<<<BROWS, 256, 0, stream>>>(gcnt, gsum, probs, sel, kRem, thr,
                                    rowSum, maskedSum, 4096, 20, 0);
  // level 1: key bits [19:8] within the selected bucket
  k_hist<<<hg, HIST_BLK, 0, stream>>>(x, gcnt, gsum, sel, 0xFFF00000u, 8, 0xFFFu);
  k_scan<<<BROWS, 256, 0, stream>>>(gcnt, gsum, probs, sel, kRem, thr,
                                    rowSum, maskedSum, 4096, 8, 1);
  // level 2: key bits [7:0] — exact threshold element + final masked-sum term
  k_hist<<<hg, HIST_BLK, 0, stream>>>(x, gcnt, gsum, sel, 0xFFFFFF00u, 0, 0xFFu);
  k_scan<<<BROWS, 256, 0, stream>>>(gcnt, gsum, probs, sel, kRem, thr,
                                    rowSum, maskedSum, 256, 0, 2);

  // scale + write (no separate masked-sum pass needed)
  k_apply<<<(BROWS * (size_t)N4PERROW) / AP_BLK, AP_BLK, 0, stream>>>(
      x, thr, rowSum, maskedSum, out);
}